// DynamicLayer_41369124995802
// MI455X (gfx1250) — compile-verified
//
#include <hip/hip_runtime.h>
#include <cstddef>

namespace {

constexpr int BB  = 4096;
constexpr int Jj  = 22;
constexpr int Dd  = 3;
constexpr int Tt  = 50;
constexpr int Nn  = 66;          // J*D
constexpr float kEps = 1e-5f;

constexpr int SST   = 52;        // padded time/K stride (50 -> 52, mult of 4)
constexpr int MROWS = 80;        // padded M rows (5 tiles of 16)
constexpr int WTS   = 64;        // padded f/N stride for transposed w_update

typedef float v2f __attribute__((ext_vector_type(2)));
typedef float v8f __attribute__((ext_vector_type(8)));

__global__ __launch_bounds__(256)
void dyn_layer(const float* __restrict__ x,
               const float* __restrict__ adj,
               const float* __restrict__ adj_t,
               const float* __restrict__ adj_jc,
               const float* __restrict__ adj_tj,
               const float* __restrict__ mlp,
               const float* __restrict__ w_update,
               const float* __restrict__ b_update,
               const float* __restrict__ alpha,
               const float* __restrict__ beta,
               const float* __restrict__ gumbels,
               float* __restrict__ out)
{
    __shared__ float xs[Nn][SST];       // input tile
    __shared__ float ss[MROWS][SST];    // s = x1 + gated term, zero-padded; reused as y
    __shared__ float wt[SST][WTS];      // wt[t][f] = w_update[f][t], zero-padded
    __shared__ float colsum[Tt];
    __shared__ float mu[Tt];
    __shared__ float rs[Tt];
    __shared__ int   gate_sh;

    const int b   = blockIdx.x;
    const int tid = threadIdx.x;
    const float* __restrict__ xb = x + (size_t)b * (Nn * Tt);

    // ---- stage x (b128 loads), transposed w_update, zero-pad ss ----
    for (int i = tid; i < (Nn * Tt) / 4; i += 256) {
        float4 v = reinterpret_cast<const float4*>(xb)[i];
        int e = i * 4;
        xs[e / Tt][e % Tt] = v.x; e++;
        xs[e / Tt][e % Tt] = v.y; e++;
        xs[e / Tt][e % Tt] = v.z; e++;
        xs[e / Tt][e % Tt] = v.w;
    }
    for (int i = tid; i < SST * WTS; i += 256) {
        int t = i / WTS, f = i % WTS;
        wt[t][f] = (t < Tt && f < Tt) ? w_update[f * Tt + t] : 0.0f;
    }
    for (int i = tid; i < (MROWS - Nn) * SST; i += 256)
        ss[Nn + i / SST][i % SST] = 0.0f;
    for (int i = tid; i < Nn * (SST - Tt); i += 256)
        ss[i / (SST - Tt)][Tt + i % (SST - Tt)] = 0.0f;
    __syncthreads();

    // ---- channel-mean column sums for the gate ----
    if (tid < Tt) {
        float s = 0.0f;
        for (int n = 0; n < Nn; ++n) s += xs[n][tid];
        colsum[tid] = s;
    }
    __syncthreads();

    // ---- gate: hard straight-through gumbel-softmax == argmax(prob+gumbel) ----
    if (tid == 0) {
        float p[4] = {0.f, 0.f, 0.f, 0.f};
        for (int t = 0; t < Tt; ++t) {
            float m = colsum[t] * (1.0f / Nn);
            p[0] += m * mlp[t * 4 + 0];
            p[1] += m * mlp[t * 4 + 1];
            p[2] += m * mlp[t * 4 + 2];
            p[3] += m * mlp[t * 4 + 3];
        }
        const float* gb = gumbels + (size_t)b * 4;
        int best = 0;
        float bv = p[0] + gb[0];          // /TAU is monotone -> skip
        for (int k = 1; k < 4; ++k) {
            float v = p[k] + gb[k];
            if (v > bv) { bv = v; best = k; }
        }
        gate_sh = best;
    }
    __syncthreads();
    const int g = gate_sh;                // uniform per block

    // ---- s = x1 (chain adjacency) + gated sparse term ----
    for (int i = tid; i < Nn * Tt; i += 256) {
        int n = i / Tt, t = i - n * Tt;
        int j = n / Dd, c = n - j * Dd;
        float v = 0.0f;
        if (j > 0)      v += adj[j * Jj + (j - 1)] * xs[n - Dd][t];
        if (j < Jj - 1) v += adj[j * Jj + (j + 1)] * xs[n + Dd][t];
        if (g == 1) {                     // shared banded temporal adjacency
            int t0 = t - 3 < 0 ? 0 : t - 3;
            int t1 = t + 3 > Tt - 1 ? Tt - 1 : t + 3;
            const float* ar = adj_t + t * Tt;
            for (int tt = t0; tt <= t1; ++tt) v += ar[tt] * xs[n][tt];
        } else if (g == 2) {              // per-joint 3x3 channel mixing
            const float* aj = adj_jc + j * 9 + c * 3;
            v += aj[0] * xs[j * Dd + 0][t];
            v += aj[1] * xs[j * Dd + 1][t];
            v += aj[2] * xs[j * Dd + 2][t];
        } else if (g == 3) {              // per-channel banded adjacency
            int t0 = t - 3 < 0 ? 0 : t - 3;
            int t1 = t + 3 > Tt - 1 ? Tt - 1 : t + 3;
            const float* ar = adj_tj + (size_t)n * Tt * Tt + t * Tt;
            for (int tt = t0; tt <= t1; ++tt) v += ar[tt] * xs[n][tt];
        }
        ss[n][t] = v;
    }
    __syncthreads();

    // ---- y = s @ w_update^T via V_WMMA_F32_16X16X4_F32 ----
    // 20 (M,N) tiles; wave w owns tn = w%4, tm in {w/4, w/4+2, w/4+4}
    const int wave = tid >> 5;
    const int lane = tid & 31;
    const int half = lane >> 4;           // K half: 0 -> K 0,1 ; 1 -> K 2,3
    const int m16  = lane & 15;
    const int tn   = wave & 3;
    const int tmb  = wave >> 2;           // 0 or 1
    const int fcol = tn * 16 + m16;
    const int r0 = tmb * 16;
    const int r1 = (tmb + 2) * 16;
    const int r2 = (tmb + 4) * 16;        // only valid when tmb==0
    // Force the wave-uniform condition into an SGPR so the third-tile WMMA
    // sits under a scalar branch (EXEC provably all-1s, no saveexec dance).
    const bool three = (__builtin_amdgcn_readfirstlane(tid) < 128);

    v8f acc0 = {}; v8f acc1 = {}; v8f acc2 = {};
    for (int ks = 0; ks < 13; ++ks) {
        const int k0 = ks * 4 + half * 2;
        v2f bf;                            // B frag reused across M tiles
        bf[0] = wt[k0][fcol];
        bf[1] = wt[k0 + 1][fcol];
        v2f af;
        af[0] = ss[r0 + m16][k0];
        af[1] = ss[r0 + m16][k0 + 1];
        acc0 = __builtin_amdgcn_wmma_f32_16x16x4_f32(false, af, false, bf,
                                                     (short)0, acc0, false, false);
        af[0] = ss[r1 + m16][k0];
        af[1] = ss[r1 + m16][k0 + 1];
        acc1 = __builtin_amdgcn_wmma_f32_16x16x4_f32(false, af, false, bf,
                                                     (short)0, acc1, false, false);
        if (three) {                       // scalar branch: EXEC untouched
            af[0] = ss[r2 + m16][k0];
            af[1] = ss[r2 + m16][k0 + 1];
            acc2 = __builtin_amdgcn_wmma_f32_16x16x4_f32(false, af, false, bf,
                                                         (short)0, acc2, false, false);
        }
    }
    __syncthreads();                       // all reads of ss done

    // ---- scatter D tiles back (reuse ss as y), add b_update ----
    if (fcol < Tt) {
        const float bu = b_update[fcol];
        #pragma unroll
        for (int r = 0; r < 8; ++r) {
            int row = r0 + r + half * 8;
            if (row < Nn) ss[row][fcol] = acc0[r] + bu;
            row = r1 + r + half * 8;
            if (row < Nn) ss[row][fcol] = acc1[r] + bu;
        }
    }
    if (three) {
        if (fcol < Tt) {
            const float bu = b_update[fcol];
            #pragma unroll
            for (int r = 0; r < 8; ++r) {
                int row = r2 + r + half * 8;
                if (row < Nn) ss[row][fcol] = acc2[r] + bu;
            }
        }
    }
    __syncthreads();

    // ---- LayerNorm over channels (axis n) ----
    if (tid < Tt) {
        float s = 0.0f;
        for (int n = 0; n < Nn; ++n) s += ss[n][tid];
        float m = s * (1.0f / Nn);
        float s2 = 0.0f;
        for (int n = 0; n < Nn; ++n) {
            float d = ss[n][tid] - m;
            s2 += d * d;
        }
        mu[tid] = m;
        rs[tid] = 1.0f / sqrtf(s2 * (1.0f / Nn) + kEps);
    }
    __syncthreads();

    // ---- out = x + normalized*alpha + beta (b128 stores) ----
    float* __restrict__ ob = out + (size_t)b * (Nn * Tt);
    for (int i = tid; i < (Nn * Tt) / 4; i += 256) {
        int e = i * 4;
        float ov[4];
        #pragma unroll
        for (int q = 0; q < 4; ++q) {
            int n = (e + q) / Tt, t = (e + q) % Tt;
            float yv = ss[n][t];
            ov[q] = xs[n][t] + (yv - mu[t]) * rs[t] * alpha[n] + beta[n];
        }
        float4 o = make_float4(ov[0], ov[1], ov[2], ov[3]);
        reinterpret_cast<float4*>(ob)[i] = o;
    }
}

} // anonymous namespace

extern "C" void kernel_launch(void* const* d_in, const int* in_sizes, int n_in,
                              void* d_out, int out_size, void* d_ws, size_t ws_size,
                              hipStream_t stream) {
    (void)in_sizes; (void)n_in; (void)out_size; (void)d_ws; (void)ws_size;
    const float* x        = (const float*)d_in[0];
    const float* adj      = (const float*)d_in[1];
    const float* adj_t    = (const float*)d_in[2];
    const float* adj_jc   = (const float*)d_in[3];
    const float* adj_tj   = (const float*)d_in[4];
    const float* mlp      = (const float*)d_in[5];
    const float* w_update = (const float*)d_in[6];
    const float* b_update = (const float*)d_in[7];
    const float* alpha    = (const float*)d_in[8];
    const float* beta     = (const float*)d_in[9];
    // d_in[10] adj_mask (chain) and d_in[11] traj_mask (band, w=3) are
    // structural and hard-coded in the kernel.
    const float* gumbels  = (const float*)d_in[12];
    float* out = (float*)d_out;

    dyn_layer<<<dim3(BB), dim3(256), 0, stream>>>(
        x, adj, adj_t, adj_jc, adj_tj, mlp, w_update, b_update,
        alpha, beta, gumbels, out);
}